// BboxLayer_79413945303921
// MI455X (gfx1250) — compile-verified
//
#include <hip/hip_runtime.h>
#include <stdint.h>

#define IBIG 0x7FFFFFFF

static constexpr int B_ = 4;
static constexpr int H_ = 384;
static constexpr int W_ = 384;
static constexpr int K_ = 64;
static constexpr int NPIX = H_ * W_;     // 147456
static constexpr int NTOT = B_ * NPIX;   // 589824

#if defined(__HIP_DEVICE_COMPILE__) && defined(__gfx1250__)
#define CDNA5_ASYNC_LDS 1
#endif

// ---------------------------------------------------------------------------
// Stage 1: threshold(textmap,linkmap) -> mask -> label init (lin idx + 1 / IBIG)
// Streams the input through LDS with CDNA5 async global->LDS copies.
// ---------------------------------------------------------------------------
__global__ __launch_bounds__(256) void k_mask_init(const float* __restrict__ x,
                                                   int* __restrict__ lab) {
  const int p = blockIdx.x * 256 + threadIdx.x;   // grid exactly covers NTOT
#if CDNA5_ASYNC_LDS
  __shared__ float tile[512];                     // 2 floats per thread
  {
    // AS3 offset == low 32 bits of the generic LDS address.
    unsigned loff = (unsigned)(uintptr_t)(&tile[threadIdx.x * 2]);
    unsigned long long ga = (unsigned long long)(uintptr_t)(x + (size_t)p * 2);
    asm volatile("global_load_async_to_lds_b64 %0, %1, off"
                 :: "v"(loff), "v"(ga) : "memory");
    asm volatile("s_wait_asynccnt 0x0" ::: "memory");
  }
  const float t = tile[threadIdx.x * 2 + 0];
  const float l = tile[threadIdx.x * 2 + 1];
#else
  const float t = x[(size_t)p * 2 + 0];
  const float l = x[(size_t)p * 2 + 1];
#endif
  const bool m = (t > 0.4f) || (l > 0.4f);        // union of the two thresholds
  const int hw = p % NPIX;                        // per-image linear index
  lab[p] = m ? (hw + 1) : IBIG;
}

// ---------------------------------------------------------------------------
// Stage 2: fill rowMinF/colMinG (contiguous region) with IBIG
// ---------------------------------------------------------------------------
__global__ __launch_bounds__(256) void k_fill(int* __restrict__ p) {
  p[blockIdx.x * 256 + threadIdx.x] = IBIG;
}

// ---------------------------------------------------------------------------
// Stage 3: connected components, min-label fixpoint.
// One workgroup per image; thread t owns row t and column t. Gauss-Seidel
// row sweeps (both directions) + column sweeps per iteration; shared
// changed-flag loop -> unique fixpoint == reference while_loop result.
// ---------------------------------------------------------------------------
__global__ __launch_bounds__(384) void k_ccl(int* __restrict__ lab) {
  int* __restrict__ L = lab + blockIdx.x * NPIX;
  const int t = threadIdx.x;
  __shared__ int sFlag;
  for (int iter = 0; iter < 4096; ++iter) {
    __syncthreads();                 // protect flag reset from prior read
    if (t == 0) sFlag = 0;
    __syncthreads();
    int ch = 0;
    {                                // horizontal sweeps over row t
      int* row = L + t * W_;
      __builtin_prefetch(row, 0, 1);               // global_prefetch_b8
      int run = IBIG;
      for (int j = 0; j < W_; ++j) {
        int v = row[j];
        if (v == IBIG)      run = IBIG;            // gap resets run
        else if (run < v) { row[j] = run; ch = 1; }
        else                run = v;
      }
      run = IBIG;
      for (int j = W_ - 1; j >= 0; --j) {
        int v = row[j];
        if (v == IBIG)      run = IBIG;
        else if (run < v) { row[j] = run; ch = 1; }
        else                run = v;
      }
    }
    __syncthreads();                 // rows visible before column sweeps
    {                                // vertical sweeps over column t
      int run = IBIG;
      for (int i = 0; i < H_; ++i) {
        int v = L[i * W_ + t];
        if (v == IBIG)      run = IBIG;
        else if (run < v) { L[i * W_ + t] = run; ch = 1; }
        else                run = v;
      }
      run = IBIG;
      for (int i = H_ - 1; i >= 0; --i) {
        int v = L[i * W_ + t];
        if (v == IBIG)      run = IBIG;
        else if (run < v) { L[i * W_ + t] = run; ch = 1; }
        else                run = v;
      }
    }
    if (ch) atomicOr(&sFlag, 1);
    __syncthreads();
    if (sFlag == 0) break;           // fixpoint reached
  }
}

// ---------------------------------------------------------------------------
// Stage 4: collect first K root labels (lab[p]==p+1) in row-major order.
// Labels equal min-linear-index+1, so scan order == ascending sorted order,
// exactly matching _component_ids (with drop beyond K). One wave32 per image.
// ---------------------------------------------------------------------------
__global__ __launch_bounds__(32) void k_ids(const int* __restrict__ lab,
                                            int* __restrict__ ids) {
  const int b = blockIdx.x;
  const int lane = threadIdx.x;
  ids[b * K_ + lane] = IBIG;          // pad slots with IBIG
  ids[b * K_ + 32 + lane] = IBIG;
  __syncthreads();
  const int* L = lab + b * NPIX;
  int cnt = 0;
  for (int base = 0; base < NPIX; base += 32) {
    const int p = base + lane;
    const int v = L[p];
    const bool root = (v == p + 1);
    const unsigned bm = (unsigned)__ballot(root);    // wave32 mask
    if (root) {
      const int pos = cnt + __popc(bm & ((1u << lane) - 1u));
      if (pos < K_) ids[b * K_ + pos] = v;
    }
    cnt += __popc(bm);
    if (cnt >= K_) break;             // uniform condition
  }
}

// ---------------------------------------------------------------------------
// Stage 5: per-(b,k): rowMinF[i] = min over comp pixels (i,j) of f(j),
//                     colMinG[j] = min over comp pixels (i,j) of g(i),
// where f(c)=c>=2?c-2:c is the leftmost column the rate-2 dilation of a pixel
// reaches (g symmetric for rows). IBIG means "row/col empty".
// ---------------------------------------------------------------------------
__global__ __launch_bounds__(256) void k_stats(const int* __restrict__ lab,
                                               const int* __restrict__ ids,
                                               int* __restrict__ rowMinF,
                                               int* __restrict__ colMinG) {
  __shared__ int sid[K_];
  const int p = blockIdx.x * 256 + threadIdx.x;
  const int b = p / NPIX;                       // 256 | NPIX, so uniform per block
  if (threadIdx.x < K_) sid[threadIdx.x] = ids[b * K_ + threadIdx.x];
  __syncthreads();
  const int v = lab[p];
  if (v == IBIG) return;                        // background
  int lo = 0, hi = K_ - 1, k = -1;              // binary search sorted ids
  while (lo <= hi) {
    const int mid = (lo + hi) >> 1;
    const int s = sid[mid];
    if (s == v) { k = mid; break; }
    if (s < v) lo = mid + 1; else hi = mid - 1;
  }
  if (k < 0) return;                            // component beyond first 64: dropped
  const int hw = p % NPIX;
  const int i = hw / W_, j = hw % W_;
  atomicMin(&rowMinF[(b * K_ + k) * H_ + i], (j >= 2) ? (j - 2) : j);
  atomicMin(&colMinG[(b * K_ + k) * W_ + j], (i >= 2) ? (i - 2) : i);
}

// ---------------------------------------------------------------------------
// Stage 6: bbox with the reference's argmax semantics.
// D row i draws from comp rows {i-2,i,i+2}; its first set column is
// min over those rows of rowMinF.  m_row[i] = nonempty && firstCol != 0.
// x2 = min i with m_row (else -1->0), w = max-min (else 1); columns symmetric.
// ---------------------------------------------------------------------------
__global__ __launch_bounds__(128) void k_bbox(const int* __restrict__ rowMinF,
                                              const int* __restrict__ colMinG,
                                              int* __restrict__ out) {
  const int bk = blockIdx.x;                    // b*K + k
  __shared__ int s_mn[2], s_mx[2];
  if (threadIdx.x == 0) { s_mn[0] = IBIG; s_mn[1] = IBIG; s_mx[0] = -1; s_mx[1] = -1; }
  __syncthreads();
  const int* rf = rowMinF + bk * H_;
  const int* cg = colMinG + bk * W_;
  for (int i = threadIdx.x; i < H_; i += 128) { // rows -> x2, w
    int m = rf[i];
    if (i >= 2)      m = min(m, rf[i - 2]);
    if (i + 2 < H_)  m = min(m, rf[i + 2]);
    if (m != IBIG && m != 0) { atomicMin(&s_mn[0], i); atomicMax(&s_mx[0], i); }
  }
  for (int j = threadIdx.x; j < W_; j += 128) { // cols -> y2, h
    int m = cg[j];
    if (j >= 2)      m = min(m, cg[j - 2]);
    if (j + 2 < W_)  m = min(m, cg[j + 2]);
    if (m != IBIG && m != 0) { atomicMin(&s_mn[1], j); atomicMax(&s_mx[1], j); }
  }
  __syncthreads();
  if (threadIdx.x == 0) {
    int x2, w, y2, h;
    if (s_mx[0] >= 0) { x2 = s_mn[0]; w = s_mx[0] - s_mn[0]; } else { x2 = 0; w = 1; }
    if (s_mx[1] >= 0) { y2 = s_mn[1]; h = s_mx[1] - s_mn[1]; } else { y2 = 0; h = 1; }
    out[bk * 4 + 0] = x2;   // where(-1 -> 0) already applied
    out[bk * 4 + 1] = y2;
    out[bk * 4 + 2] = w;
    out[bk * 4 + 3] = h;
  }
}

// ---------------------------------------------------------------------------
extern "C" void kernel_launch(void* const* d_in, const int* in_sizes, int n_in,
                              void* d_out, int out_size, void* d_ws, size_t ws_size,
                              hipStream_t stream) {
  (void)in_sizes; (void)n_in; (void)out_size; (void)ws_size;
  const float* x = (const float*)d_in[0];
  int* out = (int*)d_out;                        // reference output dtype: int32

  char* ws = (char*)d_ws;
  size_t off = 0;
  int* lab     = (int*)(ws + off); off += (size_t)NTOT * 4;          // 2.30 MB
  int* rowMinF = (int*)(ws + off); off += (size_t)B_ * K_ * H_ * 4;  // 384 KB
  int* colMinG = (int*)(ws + off); off += (size_t)B_ * K_ * W_ * 4;  // 384 KB
  int* ids     = (int*)(ws + off); off += (size_t)B_ * K_ * 4;       // 1 KB

  k_mask_init<<<NTOT / 256, 256, 0, stream>>>(x, lab);
  k_fill<<<(B_ * K_ * (H_ + W_)) / 256, 256, 0, stream>>>(rowMinF); // covers colMinG too
  k_ccl<<<B_, 384, 0, stream>>>(lab);
  k_ids<<<B_, 32, 0, stream>>>(lab, ids);
  k_stats<<<NTOT / 256, 256, 0, stream>>>(lab, ids, rowMinF, colMinG);
  k_bbox<<<B_ * K_, 128, 0, stream>>>(rowMinF, colMinG, out);
}